// ELAB_23381801959752
// MI455X (gfx1250) — compile-verified
//
#include <hip/hip_runtime.h>

typedef float v2f __attribute__((ext_vector_type(2)));
typedef float v8f __attribute__((ext_vector_type(8)));

static __device__ __forceinline__ v8f wmma_f32(v2f a, v2f b, v8f c) {
  // D = A(16x4 f32) * B(4x16 f32) + C(16x16 f32)
  return __builtin_amdgcn_wmma_f32_16x16x4_f32(false, a, false, b, (short)0, c, false, false);
}

namespace {
constexpr int BB = 4, HH = 384, WWD = 384;
constexpr int NPIX = 128;     // pixels per conv block (8 waves x 16)
constexpr int LP   = NPIX + 4;

constexpr long long S1  = (long long)BB * 60  * HH * WWD;   // 35,389,440 floats
constexpr long long S2  = (long long)BB * 120 * HH * WWD;   // 70,778,880 floats
constexpr long long NW4  = (long long)BB * (HH/4)  * (WWD/4);   // 36864
constexpr long long NW8  = (long long)BB * (HH/8)  * (WWD/8);   // 9216
constexpr long long NW12 = (long long)BB * (HH/12) * (WWD/12);  // 4096
constexpr long long A4SZ  = NW4  * 16ll  * 16ll;
constexpr long long A8SZ  = NW8  * 64ll  * 64ll;
constexpr long long A12SZ = NW12 * 144ll * 144ll;

constexpr long long OFF_SB  = 0;
constexpr long long OFF_W0  = 2048;            // 2C-channel temp (xp / lfe hidden)
constexpr long long OFF_X1  = OFF_W0 + S2;
constexpr long long OFF_X2  = OFF_X1 + S1;
constexpr long long OFF_Y   = OFF_X2 + S1;
constexpr long long OFF_A4  = OFF_Y  + S1;
constexpr long long OFF_A8  = OFF_A4 + A4SZ;
constexpr long long OFF_A12 = OFF_A8 + A8SZ;
} // namespace

// ---------------- per-channel scale/bias folding (bias, BN) ----------------
// slot layout: scale at sb[slot*256 + t], bias at sb[slot*256 + 128 + t]
__global__ void prep_sb(const float* b0, const float* b1,
                        const float* pib0, const float* g0, const float* be0,
                        const float* m0, const float* v0, const float* pob0,
                        const float* b2, const float* b3,
                        const float* pib1, const float* g1, const float* be1,
                        const float* m1, const float* v1, const float* pob1,
                        float* sb) {
  int t = threadIdx.x;
  if (t < 128) {
    for (int s = 0; s < 8; s++) { sb[s*256 + t] = 1.f; sb[s*256 + 128 + t] = 0.f; }
    if (t < 120) sb[0*256 + 128 + t] = b0[t];                        // lfe0 conv0
    if (t < 60)  sb[1*256 + 128 + t] = b1[t];                        // lfe0 conv1
    if (t < 120) {                                                   // g0 pi + BN fold
      float sc = g0[t] * rsqrtf(v0[t] + 1e-5f);
      sb[2*256 + t] = sc;
      sb[2*256 + 128 + t] = (pib0[t] - m0[t]) * sc + be0[t];
    }
    if (t < 60)  sb[3*256 + 128 + t] = pob0[t];                      // g0 po
    if (t < 120) sb[4*256 + 128 + t] = b2[t];                        // lfe1 conv0
    if (t < 60)  sb[5*256 + 128 + t] = b3[t];                        // lfe1 conv1
    if (t < 60) {                                                    // g1 pi + BN fold
      float sc = g1[t] * rsqrtf(v1[t] + 1e-5f);
      sb[6*256 + t] = sc;
      sb[6*256 + 128 + t] = (pib1[t] - m1[t]) * sc + be1[t];
    }
    if (t < 60)  sb[7*256 + 128 + t] = pob1[t];                      // g1 po
  }
}

// ---------------- fused shift5 + 1x1 conv (WMMA f32 16x16x4) ----------------
// Out[o,p] = act( scale[o]*(sum_c W[o,c]*Xshift[c,p]) + bias[o] ) [+ Res[o,p]]
template <int CIN, int COUT, int GSIZE>
__global__ __launch_bounds__(256)
void conv1x1_wmma(const float* __restrict__ X, const float* __restrict__ Wt,
                  const float* __restrict__ sb, const float* __restrict__ Res,
                  float* __restrict__ Out, int relu) {
  constexpr int MT = (COUT + 15) / 16;
  constexpr int KS = CIN / 4;
  __shared__ float Xs[CIN][LP];
  __shared__ float Wls[MT * 16][CIN + 1];
  __shared__ float Sc[128];
  __shared__ float Bi[128];

  const int tid = threadIdx.x;
  for (int e = tid; e < MT * 16 * CIN; e += 256) {
    int o = e / CIN, c = e % CIN;
    Wls[o][c] = (o < COUT) ? Wt[o * CIN + c] : 0.f;
  }
  if (tid < 128) {
    Sc[tid] = (tid < COUT) ? sb[tid] : 0.f;
    Bi[tid] = (tid < COUT) ? sb[128 + tid] : 0.f;
  }
  const long long pix0 = (long long)blockIdx.x * NPIX;
  for (int e = tid; e < CIN * NPIX; e += 256) {
    int c = e / NPIX, n = e % NPIX;
    long long p = pix0 + n;
    int b = (int)(p / (HH * WWD));
    int hw = (int)(p % (HH * WWD));
    int h = hw / WWD, w = hw % WWD;
    int dy = 0, dx = 0;
    if (GSIZE > 0) {
      constexpr int GS = (GSIZE > 0) ? GSIZE : 1;
      int g = c / GS;
      dx = (g == 0) ? 1 : ((g == 1) ? -1 : 0);
      dy = (g == 2) ? 1 : ((g == 3) ? -1 : 0);
    }
    int h2 = h + dy, w2 = w + dx;
    float v = 0.f;
    if (h2 >= 0 && h2 < HH && w2 >= 0 && w2 < WWD)
      v = X[(((long long)b * CIN + c) * HH + h2) * WWD + w2];
    Xs[c][n] = v;
  }
  __syncthreads();

  const int wv = tid >> 5, lane = tid & 31;
  const int l15 = lane & 15;
  const int khalf = (lane >= 16) ? 2 : 0;  // A/B: lanes 0-15 hold K0/K1, 16-31 hold K2/K3
  const int mhalf = (lane >= 16) ? 8 : 0;  // D: vgpr r holds M=r (lo) / M=r+8 (hi)
  const int nb = wv * 16;

  v8f acc[MT] = {};
  for (int kk = 0; kk < KS; kk++) {
    int k0 = kk * 4 + khalf;
    v2f bm; bm.x = Xs[k0][nb + l15]; bm.y = Xs[k0 + 1][nb + l15];
#pragma unroll
    for (int m = 0; m < MT; m++) {
      v2f am; am.x = Wls[m * 16 + l15][k0]; am.y = Wls[m * 16 + l15][k0 + 1];
      acc[m] = wmma_f32(am, bm, acc[m]);
    }
  }

  long long p = pix0 + nb + l15;
  int b = (int)(p / (HH * WWD));
  int hw = (int)(p % (HH * WWD));
  int h = hw / WWD, w = hw % WWD;
  const bool hasRes = (Res != nullptr);
#pragma unroll
  for (int m = 0; m < MT; m++) {
#pragma unroll
    for (int r = 0; r < 8; r++) {
      int mg = m * 16 + r + mhalf;
      if (mg < COUT) {
        float val = acc[m][r] * Sc[mg] + Bi[mg];
        if (relu) val = fmaxf(val, 0.f);
        long long idx = (((long long)b * COUT + mg) * HH + h) * WWD + w;
        if (hasRes) val += Res[idx];
        Out[idx] = val;
      }
    }
  }
}

// ---------------- windowed attention (WMMA f32 16x16x4) ----------------
// FIRST: S = q q^T, softmax -> Atn (saved), y = S v.  !FIRST: y = Atn v.
// roll(-s) before win_part and roll(+s) after win_merge cancel, so the
// gather and scatter use the same rolled coordinates.
template <int WS, bool FIRST>
__global__ __launch_bounds__((WS * WS / 16) * 32)
void attn_wmma_k(const float* __restrict__ XP, int cxp, int coff,
                 float* __restrict__ Atn, float* __restrict__ Yout, int yoff) {
  constexpr int T = WS * WS;
  constexpr int TT = T / 16;
  constexpr int CQ = 20;
  constexpr int SHIFT = WS / 2;
  __shared__ float Qs[T][24];
  __shared__ float Ss[T][T + 4];
  __shared__ float Vs[T][32];

  const int tid = threadIdx.x;
  const int nthr = TT * 32;
  const int win = blockIdx.x;
  const int wpl = WWD / WS;
  const int wpi = (HH / WS) * wpl;
  const int b = win / wpi;
  const int wrem = win % wpi;
  const int wy = wrem / wpl, wx = wrem % wpl;

  for (int e = tid; e < T * 32; e += nthr) {   // V (zero-padded to 32 cols)
    int tt = e >> 5, cv = e & 31;
    float val = 0.f;
    if (cv < CQ) {
      int ty = tt / WS, tx = tt % WS;
      int hs = (wy * WS + ty + SHIFT) % HH;
      int ws2 = (wx * WS + tx + SHIFT) % WWD;
      int c = coff + (FIRST ? (CQ + cv) : cv);
      val = XP[(((long long)b * cxp + c) * HH + hs) * WWD + ws2];
    }
    Vs[tt][cv] = val;
  }
  if (FIRST) {
    for (int e = tid; e < T * CQ; e += nthr) { // Q
      int tt = e / CQ, cq = e % CQ;
      int ty = tt / WS, tx = tt % WS;
      int hs = (wy * WS + ty + SHIFT) % HH;
      int ws2 = (wx * WS + tx + SHIFT) % WWD;
      Qs[tt][cq] = XP[(((long long)b * cxp + coff + cq) * HH + hs) * WWD + ws2];
    }
  } else {
    const float* ap = Atn + (long long)win * T * T;
    for (int e = tid; e < T * T; e += nthr) Ss[e / T][e % T] = ap[e];
  }
  __syncthreads();

  const int wv = tid >> 5, lane = tid & 31;
  const int l15 = lane & 15;
  const int khalf = (lane >= 16) ? 2 : 0;
  const int mhalf = (lane >= 16) ? 8 : 0;
  const int i = wv;  // row tile owned by this wave

  if (FIRST) {
    for (int j = 0; j < TT; j++) {
      v8f acc = {};
      for (int kk = 0; kk < CQ / 4; kk++) {
        int k0 = kk * 4 + khalf;
        v2f am; am.x = Qs[i * 16 + l15][k0]; am.y = Qs[i * 16 + l15][k0 + 1];
        v2f bm; bm.x = Qs[j * 16 + l15][k0]; bm.y = Qs[j * 16 + l15][k0 + 1];
        acc = wmma_f32(am, bm, acc);
      }
#pragma unroll
      for (int r = 0; r < 8; r++)
        Ss[i * 16 + r + mhalf][j * 16 + l15] = acc[r];
    }
    __syncthreads();
    float* ap = Atn + (long long)win * T * T;
    for (int r = tid; r < T; r += nthr) {       // row softmax + persist atn
      float mx = -3.0e38f;
      for (int c2 = 0; c2 < T; c2++) mx = fmaxf(mx, Ss[r][c2]);
      float sm = 0.f;
      for (int c2 = 0; c2 < T; c2++) { float e2 = __expf(Ss[r][c2] - mx); Ss[r][c2] = e2; sm += e2; }
      float inv = 1.f / sm;
      for (int c2 = 0; c2 < T; c2++) { float nv = Ss[r][c2] * inv; Ss[r][c2] = nv; ap[(long long)r * T + c2] = nv; }
    }
    __syncthreads();
  }

#pragma unroll
  for (int jn = 0; jn < 2; jn++) {              // y = Atn @ V (N padded to 32)
    v8f acc = {};
    for (int kk = 0; kk < T / 4; kk++) {
      int k0 = kk * 4 + khalf;
      v2f am; am.x = Ss[i * 16 + l15][k0];      am.y = Ss[i * 16 + l15][k0 + 1];
      v2f bm; bm.x = Vs[k0][jn * 16 + l15];     bm.y = Vs[k0 + 1][jn * 16 + l15];
      acc = wmma_f32(am, bm, acc);
    }
    int cv = jn * 16 + l15;
    if (cv < CQ) {
#pragma unroll
      for (int r = 0; r < 8; r++) {
        int tok = i * 16 + r + mhalf;
        int ty = tok / WS, tx = tok % WS;
        int hs = (wy * WS + ty + SHIFT) % HH;
        int ws2 = (wx * WS + tx + SHIFT) % WWD;
        Yout[(((long long)b * 60 + (yoff + cv)) * HH + hs) * WWD + ws2] = acc[r];
      }
    }
  }
}

extern "C" void kernel_launch(void* const* d_in, const int* in_sizes, int n_in,
                              void* d_out, int out_size, void* d_ws, size_t ws_size,
                              hipStream_t stream) {
  (void)in_sizes; (void)n_in; (void)out_size; (void)ws_size;
  const float* x        = (const float*)d_in[0];
  const float* lfe0_w0  = (const float*)d_in[1];
  const float* lfe0_b0  = (const float*)d_in[2];
  const float* lfe0_w1  = (const float*)d_in[3];
  const float* lfe0_b1  = (const float*)d_in[4];
  const float* g0_pi_w  = (const float*)d_in[5];
  const float* g0_pi_b  = (const float*)d_in[6];
  const float* g0_bn_g  = (const float*)d_in[7];
  const float* g0_bn_b  = (const float*)d_in[8];
  const float* g0_bn_m  = (const float*)d_in[9];
  const float* g0_bn_v  = (const float*)d_in[10];
  const float* g0_po_w  = (const float*)d_in[11];
  const float* g0_po_b  = (const float*)d_in[12];
  const float* lfe1_w0  = (const float*)d_in[13];
  const float* lfe1_b0  = (const float*)d_in[14];
  const float* lfe1_w1  = (const float*)d_in[15];
  const float* lfe1_b1  = (const float*)d_in[16];
  const float* g1_pi_w  = (const float*)d_in[17];
  const float* g1_pi_b  = (const float*)d_in[18];
  const float* g1_bn_g  = (const float*)d_in[19];
  const float* g1_bn_b  = (const float*)d_in[20];
  const float* g1_bn_m  = (const float*)d_in[21];
  const float* g1_bn_v  = (const float*)d_in[22];
  const float* g1_po_w  = (const float*)d_in[23];
  const float* g1_po_b  = (const float*)d_in[24];

  float* wsf = (float*)d_ws;
  float* SB  = wsf + OFF_SB;
  float* W0  = wsf + OFF_W0;
  float* X1  = wsf + OFF_X1;
  float* X2  = wsf + OFF_X2;
  float* Y   = wsf + OFF_Y;
  float* A4  = wsf + OFF_A4;
  float* A8  = wsf + OFF_A8;
  float* A12 = wsf + OFF_A12;
  float* Outp = (float*)d_out;

  prep_sb<<<1, 128, 0, stream>>>(lfe0_b0, lfe0_b1,
                                 g0_pi_b, g0_bn_g, g0_bn_b, g0_bn_m, g0_bn_v, g0_po_b,
                                 lfe1_b0, lfe1_b1,
                                 g1_pi_b, g1_bn_g, g1_bn_b, g1_bn_m, g1_bn_v, g1_po_b,
                                 SB);

  const int NBLK = BB * HH * WWD / NPIX;  // 4608

  // x1 = lfe0(x) + x
  conv1x1_wmma<60, 120, 12><<<NBLK, 256, 0, stream>>>(x,  lfe0_w0, SB + 0 * 256, nullptr, W0, 1);
  conv1x1_wmma<120, 60, 24><<<NBLK, 256, 0, stream>>>(W0, lfe0_w1, SB + 1 * 256, x,      X1, 0);
  // gmsa_first: xp = bn(conv_pi(x1)), window attention (atn saved), x2 = conv_po(y) + x1
  conv1x1_wmma<60, 120, 0><<<NBLK, 256, 0, stream>>>(X1, g0_pi_w, SB + 2 * 256, nullptr, W0, 0);
  attn_wmma_k<4,  true><<<(int)NW4,  32,  0, stream>>>(W0, 120, 0,  A4,  Y, 0);
  attn_wmma_k<8,  true><<<(int)NW8,  128, 0, stream>>>(W0, 120, 40, A8,  Y, 20);
  attn_wmma_k<12, true><<<(int)NW12, 288, 0, stream>>>(W0, 120, 80, A12, Y, 40);
  conv1x1_wmma<60, 60, 0><<<NBLK, 256, 0, stream>>>(Y, g0_po_w, SB + 3 * 256, X1, X2, 0);
  // x3 = lfe1(x2) + x2   (x3 stored in X1)
  conv1x1_wmma<60, 120, 12><<<NBLK, 256, 0, stream>>>(X2, lfe1_w0, SB + 4 * 256, nullptr, W0, 1);
  conv1x1_wmma<120, 60, 24><<<NBLK, 256, 0, stream>>>(W0, lfe1_w1, SB + 5 * 256, X2,     X1, 0);
  // gmsa_shared: xp = bn(conv_pi(x3)), y = atn @ v, out = conv_po(y) + x3
  conv1x1_wmma<60, 60, 0><<<NBLK, 256, 0, stream>>>(X1, g1_pi_w, SB + 6 * 256, nullptr, W0, 0);
  attn_wmma_k<4,  false><<<(int)NW4,  32,  0, stream>>>(W0, 60, 0,  A4,  Y, 0);
  attn_wmma_k<8,  false><<<(int)NW8,  128, 0, stream>>>(W0, 60, 20, A8,  Y, 20);
  attn_wmma_k<12, false><<<(int)NW12, 288, 0, stream>>>(W0, 60, 40, A12, Y, 40);
  conv1x1_wmma<60, 60, 0><<<NBLK, 256, 0, stream>>>(Y, g1_po_w, SB + 7 * 256, X1, Outp, 0);
}